// NDStack_85315230368215
// MI455X (gfx1250) — compile-verified
//
#include <hip/hip_runtime.h>
#include <hip/hip_bf16.h>

// ---------------------------------------------------------------------------
// NDStack update on gfx1250 (MI455X).
//   gamma: [B=4][16^6] fp32 (256 MB).  Memory-bound: ~256 MB of HBM traffic
//   floor (=> ~11us @ 23.3 TB/s).  All contractions done with
//   v_wmma_f32_16x16x4_f32 (native fp32 WMMA, K=4 steps).
// ---------------------------------------------------------------------------

typedef __attribute__((ext_vector_type(2))) float v2f;
typedef __attribute__((ext_vector_type(8))) float v8f;

#define WMMA_F32(a, b, c) \
    __builtin_amdgcn_wmma_f32_16x16x4_f32(false, (a), false, (b), (short)0, (c), false, false)

// Sizes (all dims are 16, B = 4)
//   gamma flat strides: a6=1, a5=16, a4=256, a3=4096, a2=65536, a1=1048576, b=16777216
//   gamma_t flat:       y=1,  r=16,  x=256,  q=4096,  i=65536,  b=1048576

// ---------------------------------------------------------------------------
// Kernel A: per batch b,  Aprev[4096 x 256] = gamma[b, row, col*16 + (t-1)]
//   new_replace[4096 x 256] = Aprev @ Rep[b]   -> written into gamma_t output
//   tmp        [4096 x 16 ] = Aprev @ Pop[b]   -> workspace
//   g8c[row,col] = gamma[b,row,col*16 + t]     -> workspace (compaction for alpha)
// One wave per 16-row tile; 1024 tiles total; 8 waves / block.
// ---------------------------------------------------------------------------
__global__ __launch_bounds__(256) void k_replace_tmp(
    const float* __restrict__ g, const float* __restrict__ rep,
    const float* __restrict__ pop, const int* __restrict__ tptr,
    float* __restrict__ gt, float* __restrict__ tmp, float* __restrict__ g8c)
{
    const int t   = *tptr;
    const int tm1 = (t > 0) ? (t - 1) : 0;
    const int lane = threadIdx.x & 31;
    const int wave = threadIdx.x >> 5;
    const int tileG = blockIdx.x * 8 + wave;      // 0..1023
    const int b  = tileG >> 8;
    const int tb = tileG & 255;                   // tile within batch
    const int M  = lane & 15;                     // A row within tile / B column
    const int hi = lane >> 4;                     // lane half
    const int K0 = hi << 1;                       // A/B k-lane offset (0 or 2)

    const int rowBase = b * 16777216 + (tb * 16 + M) * 4096;  // gamma row start
    const int g8row   = (b * 4096 + tb * 16 + M) * 256;
    const float* repB = rep + b * 65536;
    const float* popB = pop + b * 4096;

    const v8f zero = {};

    // ---- sweep 0: replace panels j = 0..7, plus g8 compaction ----
    v8f acc[8];
#pragma unroll
    for (int j = 0; j < 8; ++j) acc[j] = zero;
    for (int kk = 0; kk < 64; ++kk) {
        const int c = 4 * kk + K0;
        v2f a;
        a.x = g[rowBase + c * 16 + tm1];
        a.y = g[rowBase + (c + 1) * 16 + tm1];
        // same cache line: column t -> compacted G8 matrix
        g8c[g8row + c]     = g[rowBase + c * 16 + t];
        g8c[g8row + c + 1] = g[rowBase + (c + 1) * 16 + t];
#pragma unroll
        for (int j = 0; j < 8; ++j) {
            v2f bb;
            bb.x = repB[c * 256 + j * 16 + M];
            bb.y = repB[(c + 1) * 256 + j * 16 + M];
            acc[j] = WMMA_F32(a, bb, acc[j]);
        }
    }
    {
        float* o = gt + b * 1048576 + (tb * 16 + hi * 8) * 256 + M;
#pragma unroll
        for (int v = 0; v < 8; ++v)
#pragma unroll
            for (int j = 0; j < 8; ++j)
                o[v * 256 + j * 16] = acc[j][v];
    }

    // ---- sweep 1: replace panels j = 8..15, plus pop panel -> tmp ----
    v8f accp = zero;
#pragma unroll
    for (int j = 0; j < 8; ++j) acc[j] = zero;
    for (int kk = 0; kk < 64; ++kk) {
        const int c = 4 * kk + K0;
        v2f a;
        a.x = g[rowBase + c * 16 + tm1];
        a.y = g[rowBase + (c + 1) * 16 + tm1];
#pragma unroll
        for (int j = 0; j < 8; ++j) {
            v2f bb;
            bb.x = repB[c * 256 + (j + 8) * 16 + M];
            bb.y = repB[(c + 1) * 256 + (j + 8) * 16 + M];
            acc[j] = WMMA_F32(a, bb, acc[j]);
        }
        {
            v2f bb;
            bb.x = popB[c * 16 + M];
            bb.y = popB[(c + 1) * 16 + M];
            accp = WMMA_F32(a, bb, accp);
        }
    }
    {
        const int rowC = tb * 16 + hi * 8;
        float* o = gt + b * 1048576 + rowC * 256 + 128 + M;
#pragma unroll
        for (int v = 0; v < 8; ++v) {
#pragma unroll
            for (int j = 0; j < 8; ++j)
                o[v * 256 + j * 16] = acc[j][v];
            tmp[(b * 4096 + rowC + v) * 16 + M] = accp[v];
        }
    }
}

// ---------------------------------------------------------------------------
// Kernel B: new_pop[b,i,qx,r,y] = sum_{k=i+1..t-2} sum_u gamma[b,i,k,qx,u,y]
//                                                        * tmp[b,k,u,y,r]
// grid (rowtile=16, i=14, b=4); block = 8 waves, each wave owns 2 y values.
// Accumulate over k in registers, then += into gamma_t (holds new_replace).
// ---------------------------------------------------------------------------
__global__ __launch_bounds__(256) void k_pop(
    const float* __restrict__ g, const float* __restrict__ tmp,
    const int* __restrict__ tptr, float* __restrict__ gt)
{
    const int t = *tptr;
    const int i = blockIdx.y;
    if (i + 2 >= t) return;                      // uniform exit: need i <= t-3
    const int b    = blockIdx.z;
    const int tile = blockIdx.x;
    const int lane = threadIdx.x & 31;
    const int y    = (threadIdx.x >> 5) * 2;     // this wave: y, y+1
    const int M    = lane & 15;
    const int hi   = lane >> 4;
    const int K0   = hi << 1;

    const v8f zero = {};
    v8f acc0 = zero, acc1 = zero;

    for (int k = i + 1; k <= t - 2; ++k) {
        const float* gA = g + ((b * 16 + i) * 16 + k) * 65536 + tile * 16 * 256;
        const float* tB = tmp + (b * 4096 + k * 256) * 16;
#pragma unroll
        for (int kk = 0; kk < 4; ++kk) {
            const int u = 4 * kk + K0;
            const int abase = M * 256 + u * 16;
            v2f a0, a1, b0, b1;
            a0.x = gA[abase + y];        a0.y = gA[abase + 16 + y];
            a1.x = gA[abase + y + 1];    a1.y = gA[abase + 16 + y + 1];
            const int bbase = (u * 16 + y) * 16 + M;   // (u*16+y)*16 + r
            b0.x = tB[bbase];            b0.y = tB[bbase + 256];
            b1.x = tB[bbase + 16];       b1.y = tB[bbase + 256 + 16];
            acc0 = WMMA_F32(a0, b0, acc0);
            acc1 = WMMA_F32(a1, b1, acc1);
        }
    }

    float* o = gt + b * 1048576 + i * 65536 + (tile * 16 + hi * 8) * 256 + M * 16 + y;
#pragma unroll
    for (int v = 0; v < 8; ++v) {
        o[v * 256]     += acc0[v];
        o[v * 256 + 1] += acc1[v];
    }
}

// ---------------------------------------------------------------------------
// Kernel C: push scatter  gamma_t[b, t-1, t, x, r, y] += push[b, x, r, y, t]
// ---------------------------------------------------------------------------
__global__ __launch_bounds__(256) void k_push(
    const float* __restrict__ push, const int* __restrict__ tptr,
    float* __restrict__ gt)
{
    const int t = *tptr;
    if (t < 1 || t > 15) return;
    const int gidx = blockIdx.x * 256 + threadIdx.x;   // 0..16383
    const int b   = gidx >> 12;
    const int rem = gidx & 4095;                       // x*256 + r*16 + y
    gt[b * 1048576 + (t - 1) * 65536 + t * 4096 + rem] += push[b * 65536 + rem * 16 + t];
}

// ---------------------------------------------------------------------------
// Kernel D1: partial[b,a1,j] = sum_rr alpha[b,a1,rr] * G8[b,(a1,rr),j]
//   a1 != t -> compacted g8c;  a1 == t -> updated gamma_t[..., t]
// ---------------------------------------------------------------------------
__global__ __launch_bounds__(256) void k_alpha_partial(
    const float* __restrict__ alpha, const float* __restrict__ g8c,
    const float* __restrict__ gt, const int* __restrict__ tptr,
    float* __restrict__ partial)
{
    const int t  = *tptr;
    const int a1 = blockIdx.x;
    const int b  = blockIdx.y;
    const int j  = threadIdx.x;           // output column (r*16 + y)
    const float* al = alpha + b * 4096 + a1 * 256;
    float s = 0.0f;
    if (a1 != t) {
        const float* G = g8c + (b * 4096 + a1 * 256) * 256;
        for (int rr = 0; rr < 256; ++rr) s += al[rr] * G[rr * 256 + j];
    } else {
        const float* G = gt + b * 1048576;
        for (int rr = 0; rr < 256; ++rr) s += al[rr] * G[rr * 4096 + j * 16 + t];
    }
    partial[(b * 16 + a1) * 256 + j] = s;
}

// ---------------------------------------------------------------------------
// Kernel D2: reduce partials -> alpha_t, then obs = alpha_t / (sum + 1e-5)
// ---------------------------------------------------------------------------
__global__ __launch_bounds__(256) void k_alpha_obs(
    const float* __restrict__ partial, float* __restrict__ obs,
    float* __restrict__ alpha_t)
{
    __shared__ float red[256];
    const int b = blockIdx.x;
    const int j = threadIdx.x;
    float s = 0.0f;
    for (int c = 0; c < 16; ++c) s += partial[(b * 16 + c) * 256 + j];
    alpha_t[b * 256 + j] = s;
    red[j] = s;
    __syncthreads();
    for (int off = 128; off > 0; off >>= 1) {
        if (j < off) red[j] += red[j + off];
        __syncthreads();
    }
    obs[b * 256 + j] = s / (red[0] + 1e-5f);
}

// ---------------------------------------------------------------------------
extern "C" void kernel_launch(void* const* d_in, const int* in_sizes, int n_in,
                              void* d_out, int out_size, void* d_ws, size_t ws_size,
                              hipStream_t stream) {
    const float* gamma = (const float*)d_in[0];   // [4][16^6]
    const float* alpha = (const float*)d_in[1];   // [4][16^3]
    const float* push  = (const float*)d_in[2];   // [4][16^4]
    const float* popa  = (const float*)d_in[3];   // [4][16^3]
    const float* rep   = (const float*)d_in[4];   // [4][16^4]
    const int*   tptr  = (const int*)d_in[5];     // scalar timestep

    float* out    = (float*)d_out;
    float* obs    = out;                          // [4*256]
    float* gt     = out + 1024;                   // gamma_t  [4 * 16^5]
    float* at     = out + 1024 + 4194304;         // alpha_t  [4*256]

    float* ws      = (float*)d_ws;
    float* tmp     = ws;                          // [4*4096*16]   = 262144 f
    float* g8c     = ws + 262144;                 // [4*4096*256]  = 4194304 f
    float* partial = ws + 262144 + 4194304;       // [4*16*256]    = 16384 f

    // A: replace GEMM + pop-prep GEMM + G8 compaction (single gamma pass)
    k_replace_tmp<<<128, 256, 0, stream>>>(gamma, rep, popa, tptr, gt, tmp, g8c);
    // B: masked pop contraction, accumulated into gamma_t
    k_pop<<<dim3(16, 14, 4), 256, 0, stream>>>(gamma, tmp, tptr, gt);
    // C: push scatter
    k_push<<<64, 256, 0, stream>>>(push, tptr, gt);
    // D: alpha_t GEMV + normalization
    k_alpha_partial<<<dim3(16, 4), 256, 0, stream>>>(alpha, g8c, gt, tptr, partial);
    k_alpha_obs<<<4, 256, 0, stream>>>(partial, obs, at);
}